// Channel_86947317940845
// MI455X (gfx1250) — compile-verified
//
#include <hip/hip_runtime.h>
#include <cmath>

typedef float v2f __attribute__((ext_vector_type(2)));
typedef float v4f __attribute__((ext_vector_type(4)));
typedef float v8f __attribute__((ext_vector_type(8)));

#define LTAPS 16
#define MFFT  64
#define MK    80
#define SMAX  16
static __device__ __constant__ float kPI2 = 6.28318530717958647692f;

// D = A(16x4) x B(4x16) + C, fp32 WMMA (wave32). Chained over K.
static __device__ __forceinline__ v8f wmma_f32(v2f a, v2f b, v8f c) {
    return __builtin_amdgcn_wmma_f32_16x16x4_f32(
        /*neg_a=*/false, a, /*neg_b=*/false, b,
        /*c_mod=*/(short)0, c, /*reuse_a=*/false, /*reuse_b=*/false);
}

// Kernel 1: batched complex DFT (H_t) + Doppler weight GEMM via V_WMMA_F32_16X16X4_F32.
// Each wave handles 16 consecutive (n,p) rows (= M dim of the WMMA tile).
__global__ void __launch_bounds__(256)
dft_weight_wmma(const float* __restrict__ cof_r,
                const float* __restrict__ cof_i,
                float*  __restrict__ out_Ht,   // interleaved complex, NP*MFFT
                float2* __restrict__ weight,   // NP*S
                int NP, int S)
{
    __shared__ float2 twDFT[MFFT];          // exp(-2*pi*i*j/64)
    __shared__ float2 shiftT[SMAX * LTAPS]; // [s][l] = exp(+i*phase[l][s])

    const int tid = threadIdx.x;

    if (tid < MFFT) {
        float ang = -kPI2 * (float)tid / (float)MFFT;
        twDFT[tid] = make_float2(__cosf(ang), __sinf(ang));
    }
    for (int i = tid; i < SMAX * LTAPS; i += blockDim.x) {
        int s = i / LTAPS, l = i % LTAPS;
        float angle = kPI2 * (float)l / (float)(LTAPS - 1);
        float t     = (float)s * (0.0005f / 14.0f);
        float ph    = kPI2 * __cosf(angle) * 1000.0f * t;  // max_doppler = 1000 Hz
        shiftT[i] = (s < S) ? make_float2(__cosf(ph), __sinf(ph))
                            : make_float2(0.0f, 0.0f);
    }
    __syncthreads();

    const int lane    = tid & 31;
    const int wave    = tid >> 5;
    const int rowBase = (blockIdx.x * (int)(blockDim.x >> 5) + wave) * 16;
    if (rowBase >= NP) return;  // wave-uniform

    const int r     = lane & 15;          // A-matrix row within tile
    const int khalf = (lane >> 4) << 1;   // K sub-offset per A layout (0 or 2)
    const int mrowH = (lane >> 4) << 3;   // C/D layout: row offset 0 or 8
    const int col   = lane & 15;          // C/D layout: column = lane%16

    // Load A fragments for all 4 K-chunks (K = 16 taps total).
    v2f Ar[4], Ai[4];
    const float* cr = cof_r + (size_t)(rowBase + r) * LTAPS;
    const float* ci = cof_i + (size_t)(rowBase + r) * LTAPS;
#pragma unroll
    for (int c = 0; c < 4; ++c) {
        int k0 = 4 * c + khalf;
        Ar[c] = v2f{cr[k0], cr[k0 + 1]};
        Ai[c] = v2f{ci[k0], ci[k0 + 1]};
    }

    // ---- H_t: 4 column tiles of 16 frequencies each ----
#pragma unroll
    for (int tcol = 0; tcol < 4; ++tcol) {
        v8f accR = {}; v8f accI = {};
        const int m = tcol * 16 + col;
#pragma unroll
        for (int c = 0; c < 4; ++c) {
            int k0 = 4 * c + khalf;
            float2 t0 = twDFT[(k0 * m) & 63];
            float2 t1 = twDFT[((k0 + 1) * m) & 63];
            v2f Br  = v2f{ t0.x,  t1.x};
            v2f Bi  = v2f{ t0.y,  t1.y};
            v2f Bni = v2f{-t0.y, -t1.y};
            accR = wmma_f32(Ar[c], Br,  accR);   // Re += ar*br
            accR = wmma_f32(Ai[c], Bni, accR);   // Re -= ai*bi
            accI = wmma_f32(Ar[c], Bi,  accI);   // Im += ar*bi
            accI = wmma_f32(Ai[c], Br,  accI);   // Im += ai*br
        }
#pragma unroll
        for (int v = 0; v < 8; ++v) {
            size_t off = (size_t)(rowBase + mrowH + v) * MFFT + (size_t)m;
            v2f val = v2f{accR[v], accI[v]};
            __builtin_nontemporal_store(val, (v2f*)(out_Ht + 2 * off));
        }
    }

    // ---- weight[np][s] = sum_l cof * shift : one more 16x16 tile (cols 0..S-1 valid) ----
    {
        v8f accR = {}; v8f accI = {};
#pragma unroll
        for (int c = 0; c < 4; ++c) {
            int k0 = 4 * c + khalf;
            float2 t0 = shiftT[col * LTAPS + k0];
            float2 t1 = shiftT[col * LTAPS + k0 + 1];
            v2f Br  = v2f{ t0.x,  t1.x};
            v2f Bi  = v2f{ t0.y,  t1.y};
            v2f Bni = v2f{-t0.y, -t1.y};
            accR = wmma_f32(Ar[c], Br,  accR);
            accR = wmma_f32(Ai[c], Bni, accR);
            accI = wmma_f32(Ar[c], Bi,  accI);
            accI = wmma_f32(Ai[c], Br,  accI);
        }
        if (col < S) {
#pragma unroll
            for (int v = 0; v < 8; ++v) {
                weight[(size_t)(rowBase + mrowH + v) * S + col] =
                    make_float2(accR[v], accI[v]);
            }
        }
    }
}

// Kernel 2: pure streaming scale — out[np,s,k] = x[np,s,k] * weight[np,s].
// b128 NT loads of the real/imag planes, b128 NT stores of interleaved complex.
__global__ void __launch_bounds__(256)
apply_weight(const float*  __restrict__ xr,
             const float*  __restrict__ xi,
             const float2* __restrict__ weight,
             float* __restrict__ out, int SMK, int S)
{
    __shared__ float2 w_s[SMAX];
    const int np  = blockIdx.x;
    const int tid = threadIdx.x;
    if (tid < S) w_s[tid] = weight[(size_t)np * S + tid];
    __syncthreads();

    const v4f* xr4 = (const v4f*)(xr + (size_t)np * SMK);
    const v4f* xi4 = (const v4f*)(xi + (size_t)np * SMK);
    v4f*       o4  = (v4f*)(out + (size_t)np * SMK * 2);

    const int nvec = SMK >> 2;  // 4 complex elems per iter; MK=80 is a multiple of 4
    for (int i = tid; i < nvec; i += blockDim.x) {
        v4f a = __builtin_nontemporal_load(xr4 + i);
        v4f b = __builtin_nontemporal_load(xi4 + i);
        float2 w = w_s[(i << 2) / MK];
        v4f lo, hi;
        lo.x = a.x * w.x - b.x * w.y;  lo.y = a.x * w.y + b.x * w.x;
        lo.z = a.y * w.x - b.y * w.y;  lo.w = a.y * w.y + b.y * w.x;
        hi.x = a.z * w.x - b.z * w.y;  hi.y = a.z * w.y + b.z * w.x;
        hi.z = a.w * w.x - b.w * w.y;  hi.w = a.w * w.y + b.w * w.x;
        __builtin_nontemporal_store(lo, o4 + 2 * i);
        __builtin_nontemporal_store(hi, o4 + 2 * i + 1);
    }
}

extern "C" void kernel_launch(void* const* d_in, const int* in_sizes, int n_in,
                              void* d_out, int out_size, void* d_ws, size_t ws_size,
                              hipStream_t stream)
{
    const float* xr = (const float*)d_in[0];
    const float* xi = (const float*)d_in[1];
    const float* cr = (const float*)d_in[2];
    const float* ci = (const float*)d_in[3];

    const int NP  = in_sizes[2] / LTAPS;  // N*P = 8192
    const int SMK = in_sizes[0] / NP;     // 1120
    const int S   = SMK / MK;             // 14

    float*  out    = (float*)d_out;
    float*  out_Ht = out + (size_t)NP * SMK * 2;  // H_t follows output, both interleaved
    float2* weight = (float2*)d_ws;               // NP*S float2 (~0.9 MB scratch)

    const int rowsPerBlock = 8 * 16;  // 8 waves x 16 (n,p) rows
    const int blocks = (NP + rowsPerBlock - 1) / rowsPerBlock;
    dft_weight_wmma<<<blocks, 256, 0, stream>>>(cr, ci, out_Ht, weight, NP, S);
    apply_weight<<<NP, 256, 0, stream>>>(xr, xi, weight, out, SMK, S);
}